// BuildCost_14791867367780
// MI455X (gfx1250) — compile-verified
//
#include <hip/hip_runtime.h>

typedef __attribute__((ext_vector_type(4))) float v4f;
typedef __attribute__((ext_vector_type(4))) int   v4i;

#define HH   128
#define WW   128
#define ND   9          // disparities: -4..4
#define TILE 16         // output rows per block
#define HALO 8          // max |shift| = 4 * 2
#define LROWS (TILE + 2 * HALO)   // 32 staged input rows

#if __has_builtin(__builtin_amdgcn_global_load_async_to_lds_b128) && \
    __has_builtin(__builtin_amdgcn_s_wait_asynccnt)
#define USE_ASYNC_LDS 1
#endif

__global__ __launch_bounds__(256) void buildcost_kernel(const float* __restrict__ x,
                                                        float* __restrict__ out)
{
    __shared__ float smem[LROWS * WW];   // 16 KB

    const int tid  = threadIdx.x;
    const int img  = blockIdx.x >> 3;    // (b*c*25 + n): input image index
    const int tile = blockIdx.x & 7;
    const int t0   = tile * TILE;        // first output row of this tile

    const int n  = img % 25;
    const int a1 = n / 5;
    const int a2 = n % 5;
    const int s1 = 2 - a1;               // row shift factor
    const int s2 = 2 - a2;               // col shift factor

    const float* __restrict__ src = x + (size_t)img * (HH * WW);

    // -------- Stage 1: stage input rows [t0-8, t0+24) into LDS (zero OOB) ----
    // 32 rows * 128 floats = 1024 float4 chunks; 256 threads * 4 chunks.
#pragma unroll
    for (int k = 0; k < 4; ++k) {
        const int c    = tid + k * 256;
        const int rl   = c >> 5;              // staged row 0..31
        const int col  = (c & 31) << 2;       // column (multiple of 4)
        const int iin  = t0 - HALO + rl;      // source row in input image
        const int lofs = rl * WW + col;
        if (iin >= 0 && iin < HH) {
#ifdef USE_ASYNC_LDS
            __builtin_amdgcn_global_load_async_to_lds_b128(
                (__attribute__((address_space(1))) v4i*)(src + iin * WW + col),
                (__attribute__((address_space(3))) v4i*)(&smem[lofs]),
                /*offset=*/0, /*cpol=*/0);
#else
            *(v4f*)&smem[lofs] = *(const v4f*)(src + iin * WW + col);
#endif
        } else {
            v4f z = {0.f, 0.f, 0.f, 0.f};
            *(v4f*)&smem[lofs] = z;
        }
    }
#ifdef USE_ASYNC_LDS
    __builtin_amdgcn_s_wait_asynccnt(0);
#endif
    __syncthreads();

    // -------- Stage 2: write 9 disparity planes for this 16-row tile --------
    // 9 * 16 rows * 128 cols = 4608 float4 chunks; 256 threads * 18 chunks.
    float* __restrict__ obase = out + (size_t)img * (ND * HH * WW);
#pragma unroll
    for (int k = 0; k < 18; ++k) {
        const int e   = tid + k * 256;
        const int col = (e & 31) << 2;        // output column (multiple of 4)
        const int row = (e >> 5) & 15;        // row within tile
        const int di  = e >> 9;               // disparity index 0..8
        const int d   = di - 4;
        const int rl  = row + HALO + d * s1;  // staged LDS row, always in [0,32)
        const int dc  = d * s2;               // column shift

        const float* lrow = &smem[rl * WW];
        v4f v;
#pragma unroll
        for (int u = 0; u < 4; ++u) {
            const int js = col + u + dc;
            v[u] = (js >= 0 && js < WW) ? lrow[js] : 0.f;
        }
        float* dst = obase + (size_t)(di * HH + t0 + row) * WW + col;
        __builtin_nontemporal_store(v, (v4f*)dst);
    }
}

extern "C" void kernel_launch(void* const* d_in, const int* in_sizes, int n_in,
                              void* d_out, int out_size, void* d_ws, size_t ws_size,
                              hipStream_t stream)
{
    const float* x = (const float*)d_in[0];
    float* out = (float*)d_out;

    // images = b*c*25 ; generalizes over batch/channels
    const int images = in_sizes[0] / (HH * WW);
    const int tiles_per_img = HH / TILE;      // 8
    dim3 grid(images * tiles_per_img);
    dim3 block(256);
    buildcost_kernel<<<grid, block, 0, stream>>>(x, out);
}